// CGCA_branch_2190433321698
// MI455X (gfx1250) — compile-verified
//
#include <hip/hip_runtime.h>
#include <hip/hip_bf16.h>
#include <math.h>

typedef __attribute__((ext_vector_type(2))) float v2f;
typedef __attribute__((ext_vector_type(8))) float v8f;

#define J 17
#define CCH 512
#define CA 272
#define GRP 16
#define HW 3136
#define NEGV -9e15f

// row-major flat indices of the 49 nonzero adjacency slots (from reference ADJ)
__device__ __constant__ int NZ_IDX[49] = {
    0, 1, 17, 18, 19, 35, 36, 40, 54, 55, 57, 71, 72, 73, 89, 90,
    104, 105, 108, 109, 125, 126, 127, 143, 144, 147, 148, 152, 162, 169,
    180, 181, 197, 198, 199, 212, 215, 216, 229, 234, 235, 251, 252, 253,
    269, 270, 280, 281, 288};

// ---------------------------------------------------------------------------
// Kernel 1: per-(n,c) spatial mean over 56*56 = 3136 elements.
// One block per row; float4 streaming loads; this is the HBM-bound stage
// (411 MB read once -> ~17.6 us floor at 23.3 TB/s).
// ---------------------------------------------------------------------------
__global__ void row_mean_kernel(const float* __restrict__ x,
                                float* __restrict__ xbar) {
    const int row = blockIdx.x;                       // n*512 + c
    const float4* xr = (const float4*)(x + (size_t)row * HW);
    const int nvec = HW >> 2;                         // 784
    float s = 0.f;
    for (int i = threadIdx.x; i < nvec; i += blockDim.x) {
        float4 v = xr[i];
        s += (v.x + v.y) + (v.z + v.w);
    }
    // wave32 reduction
    for (int o = 16; o > 0; o >>= 1) s += __shfl_down(s, o, 32);
    __shared__ float wsum[8];
    const int wave = threadIdx.x >> 5;
    const int lane = threadIdx.x & 31;
    if (lane == 0) wsum[wave] = s;
    __syncthreads();
    if (threadIdx.x == 0) {
        float t = 0.f;
        const int nw = blockDim.x >> 5;
        for (int wv = 0; wv < nw; ++wv) t += wsum[wv];
        xbar[row] = t * (1.0f / (float)HW);
    }
}

// ---------------------------------------------------------------------------
// Kernel 2: x1bar[N,272] = xbar[N,512] @ w1^T[512,272] via
// V_WMMA_F32_16X16X4_F32. One wave per 16x16 output tile; 128 chained
// accumulating WMMAs over K=512.
//   A layout (16x4 f32): lanes 0-15 hold M=lane, K={k,k+1}; lanes 16-31 K={k+2,k+3}
//   B layout (4x16 f32): lanes 0-15 hold N=lane, K={k,k+1}; lanes 16-31 K={k+2,k+3}
//   D layout: VGPR r -> M = r (lanes 0-15) / r+8 (lanes 16-31), N = lane&15
// ---------------------------------------------------------------------------
__global__ void gemm_wmma_kernel(const float* __restrict__ xbar,  // [N,512]
                                 const float* __restrict__ w1,    // [272,512]
                                 float* __restrict__ x1bar,       // [N,272]
                                 int N) {
    const int tm = blockIdx.x / 17;       // M tile
    const int tn = blockIdx.x % 17;       // N tile (272 = 17*16 exactly)
    const int lane = threadIdx.x;         // 32 threads = 1 wave
    int row = tm * 16 + (lane & 15);
    if (row >= N) row = N - 1;            // clamp; OOB rows never stored
    const int col = tn * 16 + (lane & 15);
    const int koff = (lane >> 4) * 2;

    const float* ap = xbar + (size_t)row * CCH + koff;
    const float* bp = w1 + (size_t)col * CCH + koff;

    v8f acc = {};
    for (int k = 0; k < CCH; k += 4) {
        v2f a = *(const v2f*)(ap + k);    // 8-byte aligned (k+koff even)
        v2f b = *(const v2f*)(bp + k);
        acc = __builtin_amdgcn_wmma_f32_16x16x4_f32(
            /*neg_a=*/false, a, /*neg_b=*/false, b,
            /*c_mod=*/(short)0, acc, /*reuse_a=*/false, /*reuse_b=*/false);
    }

    const int mbase = tm * 16 + ((lane >> 4) ? 8 : 0);
    const int n = tn * 16 + (lane & 15);
#pragma unroll
    for (int r = 0; r < 8; ++r) {
        const int m = mbase + r;
        if (m < N) x1bar[(size_t)m * CA + n] = acc[r];
    }
}

// ---------------------------------------------------------------------------
// Kernel 3: per-batch tail. One wave per n.
//   adj = softmax(scatter(e, NZ_IDX, fill=-9e15), rows)   [17,17]
//   x2bar[ca=g*17+o] = sum_i w2[(g*17+o)*17+i] * x1bar[n, g*17+i]
//   f1[j]  = sum_ca x2bar[ca] * fc1_w[j*272+ca]
//   rgc[j] = relu(sum_k f1[k] * adj[j*17+k])
// ---------------------------------------------------------------------------
__global__ void tail1_kernel(const float* __restrict__ x1bar,  // [N,272]
                             const float* __restrict__ e,      // [49]
                             const float* __restrict__ w2,     // [272,17]
                             const float* __restrict__ fc1_w,  // [17,272]
                             float* __restrict__ rgc) {        // [N,17]
    const int n = blockIdx.x;
    const int lane = threadIdx.x;  // 32
    __shared__ float adjf[J * J];
    __shared__ float x1s[CA];
    __shared__ float x2s[CA];
    __shared__ float f1s[J];

    for (int i = lane; i < J * J; i += 32) adjf[i] = NEGV;
    __syncthreads();
    for (int i = lane; i < 49; i += 32) adjf[NZ_IDX[i]] = e[i];
    for (int i = lane; i < CA; i += 32) x1s[i] = x1bar[(size_t)n * CA + i];
    __syncthreads();

    if (lane < J) {  // softmax row `lane`
        float m = NEGV;
        for (int c = 0; c < J; ++c) m = fmaxf(m, adjf[lane * J + c]);
        float ssum = 0.f;
        float ex[J];
        for (int c = 0; c < J; ++c) {
            ex[c] = expf(adjf[lane * J + c] - m);
            ssum += ex[c];
        }
        const float inv = 1.0f / ssum;
        for (int c = 0; c < J; ++c) adjf[lane * J + c] = ex[c] * inv;
    }
    __syncthreads();

    for (int ca = lane; ca < CA; ca += 32) {  // grouped 17->17 mix
        const int g = ca / J;
        float s = 0.f;
        for (int i = 0; i < J; ++i) s += w2[ca * J + i] * x1s[g * J + i];
        x2s[ca] = s;
    }
    __syncthreads();

    if (lane < J) {  // fc1
        float s = 0.f;
        for (int ca = 0; ca < CA; ++ca) s += x2s[ca] * fc1_w[lane * CA + ca];
        f1s[lane] = s;
    }
    __syncthreads();

    if (lane < J) {  // graph matmul + relu
        float s = 0.f;
        for (int k = 0; k < J; ++k) s += f1s[k] * adjf[lane * J + k];
        rgc[(size_t)n * J + lane] = fmaxf(s, 0.f);
    }
}

// ---------------------------------------------------------------------------
// Kernel 4: out[n,c] = sigmoid(sum_j rgc[n,j] * fc2_w[c*17+j])
// ---------------------------------------------------------------------------
__global__ void tail2_kernel(const float* __restrict__ rgc,    // [N,17]
                             const float* __restrict__ fc2_w,  // [512,17]
                             float* __restrict__ out,          // [N,512]
                             int total) {
    const int idx = blockIdx.x * blockDim.x + threadIdx.x;
    if (idx >= total) return;
    const int n = idx / CCH;
    const int c = idx % CCH;
    float s = 0.f;
#pragma unroll
    for (int j = 0; j < J; ++j) s += rgc[n * J + j] * fc2_w[c * J + j];
    out[idx] = 1.0f / (1.0f + __expf(-s));
}

extern "C" void kernel_launch(void* const* d_in, const int* in_sizes, int n_in,
                              void* d_out, int out_size, void* d_ws, size_t ws_size,
                              hipStream_t stream) {
    const float* x     = (const float*)d_in[0];
    const float* e     = (const float*)d_in[1];
    const float* w1    = (const float*)d_in[2];
    const float* w2    = (const float*)d_in[3];
    const float* fc1_w = (const float*)d_in[4];
    const float* fc2_w = (const float*)d_in[5];
    float* out = (float*)d_out;

    const int N = in_sizes[0] / (CCH * HW);  // 64

    float* xbar  = (float*)d_ws;             // [N,512]
    float* x1bar = xbar + (size_t)N * CCH;   // [N,272]
    float* rgc   = x1bar + (size_t)N * CA;   // [N,17]

    row_mean_kernel<<<N * CCH, 256, 0, stream>>>(x, xbar);

    const int tilesM = (N + 15) / 16;        // 4
    gemm_wmma_kernel<<<tilesM * 17, 32, 0, stream>>>(xbar, w1, x1bar, N);

    tail1_kernel<<<N, 32, 0, stream>>>(x1bar, e, w2, fc1_w, rgc);

    const int total = N * CCH;
    tail2_kernel<<<(total + 255) / 256, 256, 0, stream>>>(rgc, fc2_w, out, total);
}